// VisionMoeBlock_69664369541779
// MI455X (gfx1250) — compile-verified
//
#include <hip/hip_runtime.h>
#include <math.h>

typedef float v2f __attribute__((ext_vector_type(2)));
typedef float v8f __attribute__((ext_vector_type(8)));

#define NBS 8
#define SL  1024
#define DIM 1024
#define FF  4096
#define NE  8

// ---------------- Kernel A: mean over sequence -> mean[b][d] ----------------
__global__ void moe_mean_kernel(const float* __restrict__ hs, float* __restrict__ mean) {
    int b = blockIdx.x;
    int d = blockIdx.y * blockDim.x + threadIdx.x;
    const float* p = hs + (size_t)b * SL * DIM + d;
    float acc = 0.f;
#pragma unroll 4
    for (int l = 0; l < SL; ++l) acc += p[(size_t)l * DIM];
    mean[b * DIM + d] = acc * (1.0f / (float)SL);
}

// ------------- Kernel B: router logits, softmax, top-2 (per batch) -----------
__global__ void moe_router_kernel(const float* __restrict__ mean,
                                  const float* __restrict__ rw,
                                  float* __restrict__ logits_out,
                                  float* __restrict__ topkw,
                                  int*   __restrict__ topki) {
    __shared__ float slog[NE];
    int b    = blockIdx.x;
    int lane = threadIdx.x & 31;
    int e    = threadIdx.x >> 5;          // 8 waves, one expert each
    const float* m = mean + b * DIM;
    const float* w = rw + e * DIM;
    float acc = 0.f;
    for (int i = lane; i < DIM; i += 32) acc += m[i] * w[i];
    for (int off = 16; off > 0; off >>= 1) acc += __shfl_down(acc, off, 32);
    if (lane == 0) slog[e] = acc;
    __syncthreads();
    if (threadIdx.x == 0) {
        float lmax = slog[0];
        for (int i = 1; i < NE; ++i) lmax = fmaxf(lmax, slog[i]);
        float p[NE]; float s = 0.f;
        for (int i = 0; i < NE; ++i) { p[i] = __expf(slog[i] - lmax); s += p[i]; }
        float inv = 1.f / s;
        for (int i = 0; i < NE; ++i) p[i] *= inv;
        int i0 = 0; float v0 = p[0];
        for (int i = 1; i < NE; ++i) if (p[i] > v0) { v0 = p[i]; i0 = i; }
        int i1 = -1; float v1 = -1.f;
        for (int i = 0; i < NE; ++i) if (i != i0 && p[i] > v1) { v1 = p[i]; i1 = i; }
        topki[2*b] = i0;  topki[2*b+1] = i1;
        topkw[2*b] = v0;  topkw[2*b+1] = v1;
        for (int i = 0; i < NE; ++i) logits_out[b * NE + i] = slog[i];
    }
}

// ---- Kernel C1: per expert, Up/Gate GEMM (M=16,N=16-tile,K=1024) + SiLU ----
// X rows: token t in 0..15 -> hidden_states[b = t>>1][l = t&1][:]
__global__ void moe_upgate_kernel(const float* __restrict__ hs,
                                  const float* __restrict__ w1,
                                  const float* __restrict__ w3,
                                  float* __restrict__ act) {
    int e    = blockIdx.x;
    int wave = threadIdx.x >> 5;
    int lane = threadIdx.x & 31;
    int f0   = (blockIdx.y * 4 + wave) * 16;
    int nm   = lane & 15;        // M index for A, N index for B/C
    int half = lane >> 4;        // K sub-pair selector
    const float* arow  = hs + (size_t)(nm >> 1) * (SL * DIM) + (size_t)(nm & 1) * DIM + 2 * half;
    const float* w1row = w1 + (size_t)e * FF * DIM + (size_t)(f0 + nm) * DIM + 2 * half;
    const float* w3row = w3 + (size_t)e * FF * DIM + (size_t)(f0 + nm) * DIM + 2 * half;
    v8f accU = {}; v8f accG = {};
    for (int k = 0; k < DIM; k += 4) {
        v2f a  = *(const v2f*)(arow  + k);
        v2f b1 = *(const v2f*)(w1row + k);
        v2f b3 = *(const v2f*)(w3row + k);
        accU = __builtin_amdgcn_wmma_f32_16x16x4_f32(false, a, false, b1, (short)0, accU, false, false);
        accG = __builtin_amdgcn_wmma_f32_16x16x4_f32(false, a, false, b3, (short)0, accG, false, false);
    }
    float* actp = act + (size_t)e * 16 * FF + f0 + nm;   // column = f0 + n
#pragma unroll
    for (int j = 0; j < 8; ++j) {
        int m = j + 8 * half;                            // C layout: VGPR j -> M=j / M=j+8
        float u = accU[j], g = accG[j];
        float silu = u / (1.f + __expf(-u));
        actp[(size_t)m * FF] = silu * g;
    }
}

// -------- Kernel C2: per expert, down GEMM (M=16, N=16-tile, K=4096) --------
__global__ void moe_down_kernel(const float* __restrict__ act,
                                const float* __restrict__ w2,
                                float* __restrict__ y) {
    int e    = blockIdx.x;
    int wave = threadIdx.x >> 5;
    int lane = threadIdx.x & 31;
    int d0   = (blockIdx.y * 4 + wave) * 16;
    int nm   = lane & 15;
    int half = lane >> 4;
    const float* arow = act + (size_t)(e * 16 + nm) * FF + 2 * half;
    const float* brow = w2  + (size_t)e * DIM * FF + (size_t)(d0 + nm) * FF + 2 * half;
    v8f acc = {};
    for (int k = 0; k < FF; k += 4) {
        v2f a  = *(const v2f*)(arow + k);
        v2f bb = *(const v2f*)(brow + k);
        acc = __builtin_amdgcn_wmma_f32_16x16x4_f32(false, a, false, bb, (short)0, acc, false, false);
    }
    float* yp = y + (size_t)e * 16 * DIM + d0 + nm;
#pragma unroll
    for (int j = 0; j < 8; ++j) {
        int m = j + 8 * half;
        yp[(size_t)m * DIM] = acc[j];
    }
}

// ------- Kernel D: weighted top-2 combine + broadcast across sequence -------
__global__ void moe_combine_kernel(const float* __restrict__ y,
                                   const float* __restrict__ topkw,
                                   const int*   __restrict__ topki,
                                   float* __restrict__ out) {
    int b = blockIdx.x;
    int d = blockIdx.y * blockDim.x + threadIdx.x;
    int e0 = topki[2*b], e1 = topki[2*b+1];
    float wa = topkw[2*b], wb = topkw[2*b+1];
    float y0 = y[(size_t)(e0 * 16 + 2*b + 0) * DIM + d];
    float y1 = y[(size_t)(e1 * 16 + 2*b + 1) * DIM + d];
    float comb = wa * y0 + wb * y1;
    float* orow = out + (size_t)b * SL * DIM + d;
    for (int l = 0; l < SL; ++l)
        __builtin_nontemporal_store(comb, orow + (size_t)l * DIM);
}

extern "C" void kernel_launch(void* const* d_in, const int* in_sizes, int n_in,
                              void* d_out, int out_size, void* d_ws, size_t ws_size,
                              hipStream_t stream) {
    const float* hs = (const float*)d_in[0];   // [8,1024,1024]
    const float* rw = (const float*)d_in[1];   // [8,1024]
    const float* w1 = (const float*)d_in[2];   // [8,4096,1024]
    const float* w2 = (const float*)d_in[3];   // [8,1024,4096]
    const float* w3 = (const float*)d_in[4];   // [8,4096,1024]
    float* out        = (float*)d_out;                       // [8,1,1024,1024]
    float* logits_out = out + (size_t)NBS * SL * DIM;        // [8,8] appended

    char* ws = (char*)d_ws;
    float* mean  = (float*)(ws);                      //  32 KB : [8,1024]
    float* topkw = (float*)(ws + 32768);              //  64 B  : [8,2]
    int*   topki = (int*)  (ws + 32832);              //  64 B  : [8,2]
    float* act   = (float*)(ws + 32896);              //   2 MB : [8,16,4096]
    float* y     = (float*)(ws + 32896 + 2097152);    // 512 KB : [8,16,1024]

    moe_mean_kernel   <<<dim3(NBS, DIM / 256), 256, 0, stream>>>(hs, mean);
    moe_router_kernel <<<dim3(NBS),            256, 0, stream>>>(mean, rw, logits_out, topkw, topki);
    moe_upgate_kernel <<<dim3(NE, FF / 64),    128, 0, stream>>>(hs, w1, w3, act);
    moe_down_kernel   <<<dim3(NE, DIM / 64),   128, 0, stream>>>(act, w2, y);
    moe_combine_kernel<<<dim3(NBS, DIM / 256), 256, 0, stream>>>(y, topkw, topki, out);
}